// RoiAlignLayer_67010079752277
// MI455X (gfx1250) — compile-verified
//
#include <hip/hip_runtime.h>
#include <cstdint>

#define POOL 7
#define CH 256

typedef __attribute__((ext_vector_type(4))) float v4f;   // native vector for NT store

// One block per ROI. 256 threads = 8 wave32s.
//   wave wv (0..7): psub = wv>>1 selects one of 4 concurrent pooled positions,
//                   half = wv&1 selects a 128-channel half.
//   lane (0..31) covers 4 channels via float4 -> each corner fetch per wave is a
//   contiguous, coalesced 512B transfer done with global_load_async_to_lds_b128.
// Double-buffered async pipeline over the 13 position-groups, synchronized with
// s_wait_asynccnt (async loads complete in order, so waiting for <=4 outstanding
// guarantees the current buffer landed while next-iteration loads are in flight).
// Output stores are non-temporal so the ~170MB of feature pyramids stay resident
// in the 192MB L2 instead of being evicted by the 100MB write-once output.
__global__ __launch_bounds__(256)
void roi_align_async_kernel(const float* __restrict__ rois,
                            const int*   __restrict__ image_shape,
                            const float* __restrict__ feat2,
                            const float* __restrict__ feat3,
                            const float* __restrict__ feat4,
                            const float* __restrict__ feat5,
                            float* __restrict__ out,
                            int B, int N)
{
    // [wave][buf][corner][lane] : 8*2*4*32*16B = 32 KB LDS
    __shared__ v4f smem[8][2][4][32];

    const int roi = blockIdx.x;            // 0 .. B*N-1
    const int b   = roi / N;

    const float y1 = rois[(size_t)roi * 4 + 0];
    const float x1 = rois[(size_t)roi * 4 + 1];
    const float y2 = rois[(size_t)roi * 4 + 2];
    const float x2 = rois[(size_t)roi * 4 + 3];
    const float h  = y2 - y1;
    const float w  = x2 - x1;

    // roi_level = clip(4 + round(log2((h*w)^2 / (224 / img_px^2))), 2, 5)
    const float img_px = (float)(image_shape[0] * image_shape[1]);
    const float denom  = 224.0f / (img_px * img_px);
    const float hw     = h * w;
    const float spec   = log2f((hw * hw) / denom);
    int lvl = 4 + (int)rintf(spec);          // rintf == round-half-even == jnp.round
    lvl = lvl < 2 ? 2 : (lvl > 5 ? 5 : lvl);

    const float* feat;
    int Hf;
    if      (lvl == 2) { feat = feat2; Hf = 256; }
    else if (lvl == 3) { feat = feat3; Hf = 128; }
    else if (lvl == 4) { feat = feat4; Hf = 64;  }
    else               { feat = feat5; Hf = 32;  }

    const size_t fbase = (size_t)b * (size_t)Hf * (size_t)Hf * CH;
    const float  scale = (float)(Hf - 1);

    const int tid   = threadIdx.x;
    const int lane  = tid & 31;
    const int wv    = tid >> 5;        // wave 0..7
    const int half  = wv & 1;          // channel half
    const int psub  = wv >> 1;         // position sub-index 0..3
    const int choff = half * 128 + lane * 4;

    // Low 32 bits of a generic LDS pointer are the LDS byte address (ISA 10.2).
    const uint32_t lds0    = (uint32_t)(uintptr_t)(&smem[wv][0][0][lane]);
    const uint32_t lds1    = (uint32_t)(uintptr_t)(&smem[wv][1][0][lane]);
    const uint32_t cstride = 32u * (uint32_t)sizeof(v4f);   // 512B between corners

    auto compute = [&](int p, uint64_t (&ga)[4], float& wx, float& wy) -> bool {
        if (p >= POOL * POOL) return false;
        const int py = p / POOL;           // constant after full unroll
        const int px = p % POOL;
        const float gyv = (float)py / 6.0f;   // folds to the exact arange/6 constant
        const float gxv = (float)px / 6.0f;
        const float ys  = (y1 + gyv * h) * scale;
        const float xs  = (x1 + gxv * w) * scale;
        const float y0f = floorf(ys);
        const float x0f = floorf(xs);
        wy = ys - y0f;
        wx = xs - x0f;
        int y0 = (int)y0f; y0 = y0 < 0 ? 0 : (y0 > Hf - 1 ? Hf - 1 : y0);
        int y1i = (y0 + 1 > Hf - 1) ? Hf - 1 : y0 + 1;
        int x0 = (int)x0f; x0 = x0 < 0 ? 0 : (x0 > Hf - 1 ? Hf - 1 : x0);
        int x1i = (x0 + 1 > Hf - 1) ? Hf - 1 : x0 + 1;
        ga[0] = (uint64_t)(uintptr_t)(feat + fbase + ((size_t)(y0  * Hf + x0 )) * CH + choff);
        ga[1] = (uint64_t)(uintptr_t)(feat + fbase + ((size_t)(y0  * Hf + x1i)) * CH + choff);
        ga[2] = (uint64_t)(uintptr_t)(feat + fbase + ((size_t)(y1i * Hf + x0 )) * CH + choff);
        ga[3] = (uint64_t)(uintptr_t)(feat + fbase + ((size_t)(y1i * Hf + x1i)) * CH + choff);
        return true;
    };

    auto issue = [&](const uint64_t (&ga)[4], uint32_t ldsbase) {
#pragma unroll
        for (int c = 0; c < 4; ++c) {
            const uint32_t la = ldsbase + (uint32_t)c * cstride;
            asm volatile("global_load_async_to_lds_b128 %0, %1, off"
                         :
                         : "v"(la), "v"(ga[c])
                         : "memory");
        }
    };

    const int NIT = (POOL * POOL + 3) / 4;   // 13 position-groups

    uint64_t gaCur[4], gaNext[4];
    float wxCur, wyCur, wxNext, wyNext;

    bool vCur = compute(psub, gaCur, wxCur, wyCur);
    if (vCur) issue(gaCur, lds0);

#pragma unroll
    for (int it = 0; it < NIT; ++it) {
        const bool vNext = compute((it + 1) * 4 + psub, gaNext, wxNext, wyNext);
        if ((it + 1 < NIT) && vNext) {
            issue(gaNext, ((it + 1) & 1) ? lds1 : lds0);
            // next-iter's 4 ops outstanding; current buffer done when <= 4 remain
            asm volatile("s_wait_asynccnt 4" ::: "memory");
        } else {
            asm volatile("s_wait_asynccnt 0" ::: "memory");
        }

        if (vCur) {
            const int buf = it & 1;
            const v4f f00 = smem[wv][buf][0][lane];
            const v4f f01 = smem[wv][buf][1][lane];
            const v4f f10 = smem[wv][buf][2][lane];
            const v4f f11 = smem[wv][buf][3][lane];

            // top = f00 + (f01-f00)*wx ; bot = f10 + (f11-f10)*wx ; o = top + (bot-top)*wy
            const v4f top = f00 + (f01 - f00) * wxCur;
            const v4f bot = f10 + (f11 - f10) * wxCur;
            const v4f o   = top + (bot - top) * wyCur;

            const int p = it * 4 + psub;
            // write-once output: non-temporal so features stay L2-resident
            __builtin_nontemporal_store(
                o, (v4f*)(out + ((size_t)roi * (POOL * POOL) + p) * CH + choff));
        }

        gaCur[0] = gaNext[0]; gaCur[1] = gaNext[1];
        gaCur[2] = gaNext[2]; gaCur[3] = gaNext[3];
        wxCur = wxNext; wyCur = wyNext;
        vCur = vNext;
    }
}

extern "C" void kernel_launch(void* const* d_in, const int* in_sizes, int n_in,
                              void* d_out, int out_size, void* d_ws, size_t ws_size,
                              hipStream_t stream) {
    const float* rois        = (const float*)d_in[0];
    const int*   image_shape = (const int*)d_in[1];
    const float* f2          = (const float*)d_in[2];
    const float* f3          = (const float*)d_in[3];
    const float* f4          = (const float*)d_in[4];
    const float* f5          = (const float*)d_in[5];
    float*       out         = (float*)d_out;

    // feat2 is (B, 256, 256, 256) -> B; rois is (B, N, 4) -> N
    int B = in_sizes[2] / (256 * 256 * 256);
    if (B < 1) B = 1;
    int N = in_sizes[0] / (4 * B);

    dim3 grid(B * N), block(256);
    hipLaunchKernelGGL(roi_align_async_kernel, grid, block, 0, stream,
                       rois, image_shape, f2, f3, f4, f5, out, B, N);
}